// Fun_42571715837961
// MI455X (gfx1250) — compile-verified
//
#include <hip/hip_runtime.h>

// CP reconstruction  out[i,j,k] = sum_p A[i,p] * B[j,p] * C[k,p]
// as GEMM  D[600 x 204800] = A[600x100] * KR^T[100x204800],
// KR[(j,k),p] = B[j,p]*C[k,p] formed on the fly per lane.
// f32 WMMA path: V_WMMA_F32_16X16X4_F32, exact f32 semantics.

typedef __attribute__((ext_vector_type(2))) float v2f;
typedef __attribute__((ext_vector_type(8))) float v8f;

#define I_DIM 600
#define J_DIM 4096
#define K_DIM 50
#define R_DIM 100
#define N_DIM (J_DIM * K_DIM)      // 204800 fused jk axis
#define NSTEPS (R_DIM / 4)         // 25 WMMA K-steps of 4
#define TILES_PER_WAVE 8           // N-tiles chained per wave (A-frag reuse)
#define WAVES_PER_BLOCK 8
#define BLOCK_THREADS (WAVES_PER_BLOCK * 32)

__global__ __launch_bounds__(BLOCK_THREADS)
void cp_wmma_f32(const float* __restrict__ A,
                 const float* __restrict__ B,
                 const float* __restrict__ C,
                 float* __restrict__ out)
{
    const int lane = threadIdx.x & 31;
    const int wv   = threadIdx.x >> 5;
    const int half = lane >> 4;        // 0: lanes 0-15, 1: lanes 16-31
    const int lm   = lane & 15;

    const int m0 = blockIdx.y * 16;

    // ---- Stage A fragments for this M-tile, all 25 K-steps (50 VGPRs) ----
    // f32 A 16x4 ISA layout: lanes 0-15 hold K=4s+0 (V0), 4s+1 (V1);
    //                        lanes 16-31 hold K=4s+2, 4s+3.
    int arow = m0 + lm;
    if (arow > I_DIM - 1) arow = I_DIM - 1;          // clamp pad rows (stores guarded)
    const v2f* aptr = (const v2f*)(A + (size_t)arow * R_DIM + 2 * half);
    v2f afrag[NSTEPS];
#pragma unroll
    for (int s = 0; s < NSTEPS; ++s)
        afrag[s] = aptr[2 * s];                      // 8B load: A[arow, 4s+2*half .. +1]

    // ---- Chain of N-tiles for this wave ----
    const int wave_n = blockIdx.x * WAVES_PER_BLOCK + wv;
    const int tbase  = wave_n * TILES_PER_WAVE;

    for (int it = 0; it < TILES_PER_WAVE; ++it) {
        const int n0 = (tbase + it) * 16;
        const int n  = n0 + lm;                      // this lane's output column
        const int j  = n / K_DIM;                    // fixed per lane per tile
        const int k  = n - j * K_DIM;
        const v2f* bp = (const v2f*)(B + (size_t)j * R_DIM + 2 * half);
        const v2f* cp = (const v2f*)(C + (size_t)k * R_DIM + 2 * half);

        v8f acc = {0.f, 0.f, 0.f, 0.f, 0.f, 0.f, 0.f, 0.f};
#pragma unroll
        for (int s = 0; s < NSTEPS; ++s) {
            // KR fragment (B-matrix 4x16): same lane-half K mapping as A.
            v2f bb = bp[2 * s];
            v2f cc = cp[2 * s];
            v2f kr;
            kr.x = bb.x * cc.x;
            kr.y = bb.y * cc.y;
            acc = __builtin_amdgcn_wmma_f32_16x16x4_f32(
                /*neg_a=*/false, afrag[s],
                /*neg_b=*/false, kr,
                /*c_mod=*/(short)0, acc,
                /*reuse_a=*/false, /*reuse_b=*/false);
        }

        // D layout (f32 16x16): VGPR v -> M = v + 8*half, N = lane&15.
        float* op = out + (size_t)n;
#pragma unroll
        for (int v = 0; v < 8; ++v) {
            int m = m0 + v + 8 * half;
            if (m < I_DIM) {
                // streaming 491MB output: non-temporal, keep L2 for B
                __builtin_nontemporal_store(acc[v], op + (size_t)m * N_DIM);
            }
        }
    }
}

extern "C" void kernel_launch(void* const* d_in, const int* in_sizes, int n_in,
                              void* d_out, int out_size, void* d_ws, size_t ws_size,
                              hipStream_t stream) {
    const float* A = (const float*)d_in[0];   // (600, 100)
    const float* B = (const float*)d_in[1];   // (4096, 100)
    const float* C = (const float*)d_in[2];   // (50, 100)
    float* out = (float*)d_out;               // (600, 4096, 50) f32

    // N-tiles: 204800/16 = 12800 = gridDim.x * WAVES_PER_BLOCK * TILES_PER_WAVE
    dim3 grid(N_DIM / (16 * TILES_PER_WAVE * WAVES_PER_BLOCK),   // 200
              (I_DIM + 15) / 16);                                // 38
    cp_wmma_f32<<<grid, BLOCK_THREADS, 0, stream>>>(A, B, C, out);
}